// BinaryTreeRNN_35493609734250
// MI455X (gfx1250) — compile-verified
//
#include <hip/hip_runtime.h>

typedef __attribute__((ext_vector_type(2))) float v2f;
typedef __attribute__((ext_vector_type(8))) float v8f;

// Folded-parameter layout in d_ws (floats)
#define P_OW3 0   // 16: w3[n]*softmax(om3[n]) (4 nodes x 4 ops)
#define P_B3  16  // 4
#define P_OW2 20  // 8:  w2[n]*softmax(om2[n]) (2 nodes x 4 ops)
#define P_B2  28  // 2
#define P_OW1 30  // 4:  w1*softmax(om1)
#define P_B1  34  // 1

__global__ void prep_params_kernel(const float* __restrict__ w1, const float* __restrict__ b1,
                                   const float* __restrict__ om1,
                                   const float* __restrict__ w2, const float* __restrict__ b2,
                                   const float* __restrict__ om2,
                                   const float* __restrict__ w3, const float* __restrict__ b3,
                                   const float* __restrict__ om3,
                                   float* __restrict__ P)
{
    if (threadIdx.x != 0 || blockIdx.x != 0) return;
    // level 3: 4 nodes
    for (int n = 0; n < 4; ++n) {
        float mx = om3[n*4+0];
        for (int o = 1; o < 4; ++o) mx = fmaxf(mx, om3[n*4+o]);
        float e[4]; float s = 0.f;
        for (int o = 0; o < 4; ++o) { e[o] = __expf(om3[n*4+o] - mx); s += e[o]; }
        float sc = w3[n] / s;
        for (int o = 0; o < 4; ++o) P[P_OW3 + n*4 + o] = e[o] * sc;
        P[P_B3 + n] = b3[n];
    }
    // level 2: 2 nodes
    for (int n = 0; n < 2; ++n) {
        float mx = om2[n*4+0];
        for (int o = 1; o < 4; ++o) mx = fmaxf(mx, om2[n*4+o]);
        float e[4]; float s = 0.f;
        for (int o = 0; o < 4; ++o) { e[o] = __expf(om2[n*4+o] - mx); s += e[o]; }
        float sc = w2[n] / s;
        for (int o = 0; o < 4; ++o) P[P_OW2 + n*4 + o] = e[o] * sc;
        P[P_B2 + n] = b2[n];
    }
    // level 1: 1 node
    {
        float mx = om1[0];
        for (int o = 1; o < 4; ++o) mx = fmaxf(mx, om1[o]);
        float e[4]; float s = 0.f;
        for (int o = 0; o < 4; ++o) { e[o] = __expf(om1[o] - mx); s += e[o]; }
        float sc = w1[0] / s;
        for (int o = 0; o < 4; ++o) P[P_OW1 + o] = e[o] * sc;
        P[P_B1] = b1[0];
    }
}

__device__ __forceinline__ float node_eval(float l, float r, const float* ow, float b)
{
    float s  = l + r;
    float p  = l * r;
    float sn = __sinf(s);
    float cs = __cosf(s);
    return fmaf(ow[0], s, fmaf(ow[1], sn, fmaf(ow[2], cs, fmaf(ow[3], p, b))));
}

__global__ __launch_bounds__(256)
void tree_wmma_kernel(const float* __restrict__ x,
                      const float* __restrict__ leaf_w,
                      const float* __restrict__ leaf_b,
                      const float* __restrict__ P,
                      float* __restrict__ out, int nrows)
{
    const int lane = threadIdx.x & 31;
    const int m16  = lane & 15;
    const int hi   = lane >> 4;   // 0: lanes 0-15, 1: lanes 16-31

    const long long wavesPerBlock = blockDim.x >> 5;
    const long long wid       = (long long)blockIdx.x * wavesPerBlock + (threadIdx.x >> 5);
    const long long waveCount = (long long)gridDim.x * wavesPerBlock;

    // ---- A fragments: leaf_w (8x16) zero-padded to 16x16, used as A for all tiles.
    // A 16x4 f32 layout: lane m (and m+16) holds row m; VGPR0 = K{2*hi}, VGPR1 = K{2*hi+1}.
    v2f a[4];
    {
        const bool valid = (m16 < 8);
        #pragma unroll
        for (int s = 0; s < 4; ++s) {
            const int k0 = 4*s + 2*hi;
            a[s].x = valid ? leaf_w[m16*16 + k0]     : 0.f;
            a[s].y = valid ? leaf_w[m16*16 + k0 + 1] : 0.f;
        }
    }

    float lb[8];
    #pragma unroll
    for (int r = 0; r < 8; ++r) lb[r] = leaf_b[r];

    float ow3[16], b3v[4], ow2[8], b2v[2], ow1v[4], b1v;
    #pragma unroll
    for (int i = 0; i < 16; ++i) ow3[i] = P[P_OW3 + i];
    #pragma unroll
    for (int i = 0; i < 4;  ++i) b3v[i] = P[P_B3 + i];
    #pragma unroll
    for (int i = 0; i < 8;  ++i) ow2[i] = P[P_OW2 + i];
    #pragma unroll
    for (int i = 0; i < 2;  ++i) b2v[i] = P[P_B2 + i];
    #pragma unroll
    for (int i = 0; i < 4;  ++i) ow1v[i] = P[P_OW1 + i];
    b1v = P[P_B1];

    // ---- main loop: each wave consumes 32 rows (two 16-row WMMA tiles) per iteration.
    for (long long base = wid * 32; base + 32 <= (long long)nrows; base += waveCount * 32) {
        // B 4x16 f32 layout: lane l holds B[2*hi + v][m16] in VGPR v
        //   => lane needs x[base + m16][4*s + 2*hi + {0,1}]  (8-byte coalesced loads)
        const float* r0 = x + (base + m16) * 16 + 2*hi;
        const float* r1 = r0 + 256;   // next 16 rows

        v8f d0 = {0.f,0.f,0.f,0.f,0.f,0.f,0.f,0.f};
        v8f d1 = {0.f,0.f,0.f,0.f,0.f,0.f,0.f,0.f};
        #pragma unroll
        for (int s = 0; s < 4; ++s) {
            v2f bf = __builtin_nontemporal_load((const v2f*)(r0 + 4*s));
            d0 = __builtin_amdgcn_wmma_f32_16x16x4_f32(false, a[s], false, bf,
                                                       (short)0, d0, false, false);
        }
        #pragma unroll
        for (int s = 0; s < 4; ++s) {
            v2f bf = __builtin_nontemporal_load((const v2f*)(r1 + 4*s));
            d1 = __builtin_amdgcn_wmma_f32_16x16x4_f32(false, a[s], false, bf,
                                                       (short)0, d1, false, false);
        }

        // D layout: VGPR r, lanes 0-15 hold h_r(row base+lane). Merge tile1 into lanes 16-31.
        float h[8];
        #pragma unroll
        for (int r = 0; r < 8; ++r) {
            float other = __shfl(d1[r], m16, 32);
            h[r] = (hi ? other : d0[r]) + lb[r];
        }

        // lane-local binary tree
        float g0 = node_eval(h[0], h[1], &ow3[0],  b3v[0]);
        float g1 = node_eval(h[2], h[3], &ow3[4],  b3v[1]);
        float g2 = node_eval(h[4], h[5], &ow3[8],  b3v[2]);
        float g3 = node_eval(h[6], h[7], &ow3[12], b3v[3]);
        float f0 = node_eval(g0, g1, &ow2[0], b2v[0]);
        float f1 = node_eval(g2, g3, &ow2[4], b2v[1]);
        float res = node_eval(f0, f1, ow1v, b1v);

        __builtin_nontemporal_store(res, out + base + lane);
    }

    // ---- scalar tail for nrows % 32 (not hit for N = 4,000,000, kept for generality)
    const int tailStart = nrows & ~31;
    const int tail      = nrows - tailStart;
    if (tail > 0) {
        const int g = blockIdx.x * blockDim.x + threadIdx.x;
        if (g < tail) {
            const long long row = tailStart + g;
            float h[8];
            #pragma unroll
            for (int r = 0; r < 8; ++r) {
                float acc = lb[r];
                #pragma unroll
                for (int k = 0; k < 16; ++k)
                    acc = fmaf(leaf_w[r*16 + k], x[row*16 + k], acc);
                h[r] = acc;
            }
            float g0 = node_eval(h[0], h[1], &ow3[0],  b3v[0]);
            float g1 = node_eval(h[2], h[3], &ow3[4],  b3v[1]);
            float g2 = node_eval(h[4], h[5], &ow3[8],  b3v[2]);
            float g3 = node_eval(h[6], h[7], &ow3[12], b3v[3]);
            float f0 = node_eval(g0, g1, &ow2[0], b2v[0]);
            float f1 = node_eval(g2, g3, &ow2[4], b2v[1]);
            out[row] = node_eval(f0, f1, ow1v, b1v);
        }
    }
}

extern "C" void kernel_launch(void* const* d_in, const int* in_sizes, int n_in,
                              void* d_out, int out_size, void* d_ws, size_t ws_size,
                              hipStream_t stream)
{
    const float* x      = (const float*)d_in[0];
    const float* leaf_w = (const float*)d_in[1];
    const float* leaf_b = (const float*)d_in[2];
    const float* w1     = (const float*)d_in[3];
    const float* b1     = (const float*)d_in[4];
    const float* om1    = (const float*)d_in[5];
    const float* w2     = (const float*)d_in[6];
    const float* b2     = (const float*)d_in[7];
    const float* om2    = (const float*)d_in[8];
    const float* w3     = (const float*)d_in[9];
    const float* b3     = (const float*)d_in[10];
    const float* om3    = (const float*)d_in[11];
    float* out = (float*)d_out;
    float* P   = (float*)d_ws;
    const int nrows = out_size;   // N

    prep_params_kernel<<<1, 32, 0, stream>>>(w1, b1, om1, w2, b2, om2, w3, b3, om3, P);

    const int threads = 256;      // 8 waves/block
    const int blocks  = 4096;     // 32768 waves, grid-stride over 125000 wave-tiles
    tree_wmma_kernel<<<blocks, threads, 0, stream>>>(x, leaf_w, leaf_b, P, out, nrows);
}